// GraphConvolution_37941741093302
// MI455X (gfx1250) — compile-verified
//
#include <hip/hip_runtime.h>
#include <hip/hip_bf16.h>

typedef __attribute__((ext_vector_type(2))) float v2f;
typedef __attribute__((ext_vector_type(8))) float v8f;

#define IN_FEAT   256
#define OUT_FEAT  64
#define LDS_PITCH 258   // floats; odd-ish pitch spreads LDS banks

// ---------------------------------------------------------------------------
// Kernel 1: h = x @ W via V_WMMA_F32_16X16X4_F32.
// Block = 256 threads = 8 waves. Block tile: 64 rows x 32 cols.
//   wave w: row-tile mt = w>>1 (0..3), col-tile nt = w&1 (0..1).
// W column slab [256 x 32] staged transposed in LDS (pitch 258 floats).
// ---------------------------------------------------------------------------
__global__ __launch_bounds__(256) void gc_gemm_wmma(
    const float* __restrict__ x, const float* __restrict__ w,
    float* __restrict__ h, int n_nodes)
{
  __shared__ float wt[32 * LDS_PITCH];          // 33,024 bytes

  const int cbase = blockIdx.y * 32;            // which 32 output cols
  // Cooperative transposed load of W[0:256][cbase:cbase+32] -> wt[c][k]
  for (int i = threadIdx.x; i < 256 * 32; i += 256) {
    int r = i >> 5;                             // K index 0..255
    int cl = i & 31;                            // local col 0..31
    wt[cl * LDS_PITCH + r] = w[r * OUT_FEAT + cbase + cl];
  }
  __syncthreads();

  const int wid  = threadIdx.x >> 5;
  const int lane = threadIdx.x & 31;
  const int mt   = wid >> 1;                    // 0..3
  const int nt   = wid & 1;                     // 0..1
  const int row0 = blockIdx.x * 64 + mt * 16;
  const int n0l  = nt * 16;                     // local col base in slab

  const int m  = lane & 15;                     // M (A) / N (B) index
  const int hi = lane >> 4;                     // K sub-pair select

  int arow = row0 + m;
  if (arow >= n_nodes) arow = n_nodes - 1;      // clamp loads; EXEC stays full
  const float* xrow = x + (size_t)arow * IN_FEAT + 2 * hi;
  const float* bcol = &wt[(n0l + m) * LDS_PITCH + 2 * hi];

  v8f acc = {};
  #pragma unroll 8
  for (int k = 0; k < IN_FEAT; k += 4) {
    v2f a = *(const v2f*)(xrow + k);            // {x[m][k+2hi], x[m][k+2hi+1]}
    v2f b = *(const v2f*)(bcol + k);            // {W[k+2hi][n], W[k+2hi+1][n]}
    acc = __builtin_amdgcn_wmma_f32_16x16x4_f32(
        /*neg_a=*/false, a, /*neg_b=*/false, b,
        /*c_mod=*/(short)0, acc, /*reuse_a=*/false, /*reuse_b=*/false);
  }

  // D layout: VGPR r -> row (r + 8*hi), col (n0 + m)
  const int nOut = cbase + n0l + m;
  #pragma unroll
  for (int r = 0; r < 8; ++r) {
    int mr = row0 + r + 8 * hi;
    if (mr < n_nodes) h[(size_t)mr * OUT_FEAT + nOut] = acc[r];
  }
}

// ---------------------------------------------------------------------------
// Kernel 2: zero the accumulator (d_out is poisoned by the harness).
// ---------------------------------------------------------------------------
__global__ __launch_bounds__(256) void gc_zero(float4* __restrict__ p, int n4)
{
  int i = blockIdx.x * 256 + threadIdx.x;
  if (i < n4) p[i] = make_float4(0.f, 0.f, 0.f, 0.f);
}

// ---------------------------------------------------------------------------
// Kernel 3: SpMM scatter. One wave per edge; lane handles features
// {2*lane, 2*lane+1} -> one float2 gather from h + two f32 global atomics.
// h (25.6 MB) and agg (25.6 MB) both resident in the 192 MB L2.
// ---------------------------------------------------------------------------
__global__ __launch_bounds__(256) void gc_spmm(
    const float* __restrict__ h, const int* __restrict__ rows,
    const int* __restrict__ cols, const float* __restrict__ vals,
    float* __restrict__ agg, int n_edges)
{
  int e = blockIdx.x * 8 + (threadIdx.x >> 5);
  if (e >= n_edges) return;
  int lane = threadIdx.x & 31;
  int r = rows[e];
  int c = cols[e];
  float v = vals[e];
  float2 hv = ((const float2*)(h + (size_t)c * OUT_FEAT))[lane];
  float* dst = agg + (size_t)r * OUT_FEAT + 2 * lane;
  atomicAdd(dst,     v * hv.x);
  atomicAdd(dst + 1, v * hv.y);
}

// ---------------------------------------------------------------------------
// Kernel 4: in-place row softmax (64 wide). One wave per node, wave32
// shuffle reductions for max and sum.
// ---------------------------------------------------------------------------
__global__ __launch_bounds__(256) void gc_softmax(float* __restrict__ io,
                                                  int n_nodes)
{
  int node = blockIdx.x * 8 + (threadIdx.x >> 5);
  if (node >= n_nodes) return;
  int lane = threadIdx.x & 31;
  float2 v = ((const float2*)(io + (size_t)node * OUT_FEAT))[lane];

  float m = fmaxf(v.x, v.y);
  #pragma unroll
  for (int off = 16; off >= 1; off >>= 1)
    m = fmaxf(m, __shfl_xor(m, off, 32));

  float ex = __expf(v.x - m);
  float ey = __expf(v.y - m);
  float s = ex + ey;
  #pragma unroll
  for (int off = 16; off >= 1; off >>= 1)
    s += __shfl_xor(s, off, 32);

  float inv = 1.0f / s;
  ((float2*)(io + (size_t)node * OUT_FEAT))[lane] =
      make_float2(ex * inv, ey * inv);
}

// ---------------------------------------------------------------------------
extern "C" void kernel_launch(void* const* d_in, const int* in_sizes, int n_in,
                              void* d_out, int out_size, void* d_ws, size_t ws_size,
                              hipStream_t stream)
{
  const float* x  = (const float*)d_in[0];   // [N, 256]
  const float* w  = (const float*)d_in[1];   // [256, 64]
  const int*   er = (const int*)  d_in[2];   // [E]
  const int*   ec = (const int*)  d_in[3];   // [E]
  const float* ev = (const float*)d_in[4];   // [E]
  float* out = (float*)d_out;                // [N, 64] (accumulator + result)

  const int n_nodes = in_sizes[0] / IN_FEAT;
  const int n_edges = in_sizes[2];

  float* h = (float*)d_ws;                   // [N, 64] projection scratch

  // 1) dense projection (WMMA f32)
  dim3 ggrid((n_nodes + 63) / 64, 2);
  gc_gemm_wmma<<<ggrid, 256, 0, stream>>>(x, w, h, n_nodes);

  // 2) zero accumulator
  int n4 = out_size / 4;
  gc_zero<<<(n4 + 255) / 256, 256, 0, stream>>>((float4*)out, n4);

  // 3) edge scatter with f32 atomics
  gc_spmm<<<(n_edges + 7) / 8, 256, 0, stream>>>(h, er, ec, ev, out, n_edges);

  // 4) in-place softmax
  gc_softmax<<<(n_nodes + 7) / 8, 256, 0, stream>>>(out, n_nodes);
}